// TopKGate_83416854823545
// MI455X (gfx1250) — compile-verified
//
#include <hip/hip_runtime.h>
#include <hip/hip_bf16.h>

typedef __attribute__((ext_vector_type(2))) float v2f;
typedef __attribute__((ext_vector_type(4))) float v4f;
typedef __attribute__((ext_vector_type(8))) float v8f;

#define TOK   4096
#define NEXP  8
#define KDIM  1024

// ---------------------------------------------------------------------------
// Kernel 1: gate logits via V_WMMA_F32_16X16X4_F32, softmax over E=8,
// per-block entropy partial. One wave (32 threads) per 16-token tile.
// A (16x4 f32): lanes 0-15 -> M=lane, v0=K(k+0), v1=K(k+1); lanes 16-31 -> K+2/K+3
// B (4x16 f32): mirrored, N=lane&15. Lanes with N>=8 alias expert N&7 and are
// zeroed with a multiply -> no divergent EXEC masking in the inner loop.
// D (16x16 f32): vgpr r -> M=r (lanes 0-15), M=8+r (lanes 16-31), N=lane&15
// ---------------------------------------------------------------------------
__global__ __launch_bounds__(32) void gates_wmma_kernel(
    const float* __restrict__ inp, const float* __restrict__ wg,
    float* __restrict__ gates, float* __restrict__ partials) {
  const int tile = blockIdx.x;        // 0..255
  const int lane = threadIdx.x;       // 0..31
  const int half = lane >> 4;         // 0/1
  const int l15  = lane & 15;

  __shared__ float lds_logits[16 * 16];
  __shared__ float lds_h[16];

  const int row = tile * 16 + l15;                      // token row of A
  const float* arow = inp + (size_t)row * KDIM + 2 * half;
  const float* brow = wg + (size_t)(l15 & 7) * KDIM + 2 * half; // clamped expert
  const float bscale = (l15 < NEXP) ? 1.0f : 0.0f;      // zero pad lanes

  v8f c = {};
  #pragma unroll 4
  for (int k = 0; k < KDIM; k += 4) {
    v2f a = *(const v2f*)(arow + k);                    // A[row][k+2h : k+2h+1]
    v2f b = *(const v2f*)(brow + k);                    // B[k+2h][n] = wg[n][k+2h]
    b.x *= bscale;
    b.y *= bscale;
    // 8 args: (neg_a, A, neg_b, B, c_mod, C, reuse_a, reuse_b)
    c = __builtin_amdgcn_wmma_f32_16x16x4_f32(
        false, a, false, b, (short)0, c, false, false);
  }

  // spill 16x16 logits to LDS: element (M = r + 8*half, N = l15)
  #pragma unroll
  for (int r = 0; r < 8; ++r)
    lds_logits[(r + 8 * half) * 16 + l15] = c[r];
  __syncthreads();

  // 16 lanes: one token each -> softmax over 8 experts + entropy term
  if (lane < 16) {
    const float* lg = lds_logits + lane * 16;
    float m = lg[0];
    #pragma unroll
    for (int e = 1; e < NEXP; ++e) m = fmaxf(m, lg[e]);
    float p[NEXP];
    float s = 0.0f;
    #pragma unroll
    for (int e = 0; e < NEXP; ++e) { p[e] = expf(lg[e] - m); s += p[e]; }
    const float inv = 1.0f / s;
    const int token = tile * 16 + lane;
    float h = 0.0f;
    #pragma unroll
    for (int e = 0; e < NEXP; ++e) {
      float g = p[e] * inv;
      gates[token * NEXP + e] = g;
      h -= g * logf(fmaxf(g, 1e-38f));
    }
    lds_h[lane] = h;
  }
  __syncthreads();
  if (lane == 0) {
    float s = 0.0f;
    #pragma unroll
    for (int i = 0; i < 16; ++i) s += lds_h[i];
    partials[tile] = s;
  }
}

// ---------------------------------------------------------------------------
// Kernel 2: streaming zero-fill of combine_sec + dispatch_mask (1.07 GB)
// with non-temporal 128-bit stores (write-once, bypass cache retention).
// This is the roofline-dominant kernel: ~46 us at 23.3 TB/s.
// ---------------------------------------------------------------------------
__global__ __launch_bounds__(256) void zero_fill_kernel(v4f* __restrict__ out,
                                                        long long n4) {
  long long i = (long long)blockIdx.x * blockDim.x + threadIdx.x;
  const long long stride = (long long)gridDim.x * blockDim.x;
  v4f z = {0.0f, 0.0f, 0.0f, 0.0f};
  for (; i < n4; i += stride)
    __builtin_nontemporal_store(z, out + i);
}

// ---------------------------------------------------------------------------
// Kernel 3: place diagonal: combine_sec[t,e,t] = gates[t,e]; mask diag = 1.0
// ---------------------------------------------------------------------------
__global__ __launch_bounds__(256) void diag_scatter_kernel(
    const float* __restrict__ gates, float* __restrict__ out) {
  const int idx = blockIdx.x * blockDim.x + threadIdx.x;  // 0..32767
  const int t = idx >> 3;
  const int e = idx & 7;
  const size_t off = (size_t)t * (NEXP * TOK) + (size_t)e * TOK + (size_t)t;
  const size_t D = (size_t)TOK * NEXP * TOK;
  out[off]     = gates[idx];   // combine_sec diagonal
  out[D + off] = 1.0f;         // dispatch_mask diagonal (gates > 0 always)
}

// ---------------------------------------------------------------------------
// Kernel 4: reduce 256 entropy partials -> mean entropy scalar
// ---------------------------------------------------------------------------
__global__ __launch_bounds__(256) void entropy_finalize_kernel(
    const float* __restrict__ partials, float* __restrict__ out_entropy) {
  __shared__ float s[256];
  s[threadIdx.x] = partials[threadIdx.x];
  __syncthreads();
  for (int off = 128; off > 0; off >>= 1) {
    if ((int)threadIdx.x < off) s[threadIdx.x] += s[threadIdx.x + off];
    __syncthreads();
  }
  if (threadIdx.x == 0) *out_entropy = s[0] / (float)TOK;
}

// ---------------------------------------------------------------------------
extern "C" void kernel_launch(void* const* d_in, const int* in_sizes, int n_in,
                              void* d_out, int out_size, void* d_ws, size_t ws_size,
                              hipStream_t stream) {
  (void)in_sizes; (void)n_in; (void)out_size; (void)ws_size;
  const float* inp = (const float*)d_in[0];   // [4096, 1024] f32
  const float* wg  = (const float*)d_in[1];   // [8, 1024] f32
  float* out = (float*)d_out;                 // combine_sec | mask | entropy

  float* gates    = (float*)d_ws;             // 32768 floats
  float* partials = gates + TOK * NEXP;       // 256 floats

  // 1) gates + entropy partials (256 tiles x 1 wave)
  gates_wmma_kernel<<<256, 32, 0, stream>>>(inp, wg, gates, partials);

  // 2) zero-fill both big outputs: 2*T*E*T floats = 67,108,864 float4
  const long long n4 = (long long)2 * TOK * NEXP * TOK / 4;
  zero_fill_kernel<<<8192, 256, 0, stream>>>((v4f*)out, n4);

  // 3) diagonal scatter (depends on gates + fill; stream-ordered)
  diag_scatter_kernel<<<(TOK * NEXP) / 256, 256, 0, stream>>>(gates, out);

  // 4) entropy scalar at out[2*T*E*T]
  entropy_finalize_kernel<<<1, 256, 0, stream>>>(
      partials, out + (size_t)2 * TOK * NEXP * TOK);
}